// disentangle_46969762349144
// MI455X (gfx1250) — compile-verified
//
#include <hip/hip_runtime.h>
#include <stdint.h>

// out[r,i] = x[r,i] + ordinal_rank(|x[r,:]|)[i] * sign(x[r,i]) / (L-1)
// One workgroup per row; row DMA'd to LDS via the CDNA5 Tensor Data Mover,
// ranked with an in-LDS bitonic sort of (absbits<<32 | index) keys.

#define LDIM    4096
#define THREADS 1024

typedef __attribute__((ext_vector_type(4))) unsigned int u32x4;
typedef __attribute__((ext_vector_type(8))) int          i32x8;
typedef __attribute__((ext_vector_type(4))) int          i32x4;

__global__ __launch_bounds__(THREADS)
void rank_code_kernel(const float* __restrict__ x, float* __restrict__ out, int nrows) {
    __shared__ float          lds_x[LDIM];      // 16 KB: raw row
    __shared__ uint64_t       lds_key[LDIM];    // 32 KB: sort keys
    __shared__ unsigned short lds_rank[LDIM];   //  8 KB: rank per original index

    const int row = blockIdx.x;
    if (row >= nrows) return;
    const float* src = x + (size_t)row * LDIM;

    // ---- Stage 1: bring the 16KB row into LDS ------------------------------
#if __has_builtin(__builtin_amdgcn_tensor_load_to_lds) && __has_builtin(__builtin_amdgcn_s_wait_tensorcnt)
    if (threadIdx.x < 32) {  // one wave issues the TDM DMA (EXEC-independent op)
        const uint64_t g       = (uint64_t)(uintptr_t)src;
        const uint32_t lds_off = (uint32_t)(uintptr_t)(&lds_x[0]);

        // D# group 0: count=1, lds_addr, global_addr[56:0], type=2 ("image")
        u32x4 g0;
        g0[0] = 1u;
        g0[1] = lds_off;
        g0[2] = (uint32_t)(g & 0xFFFFFFFFull);
        g0[3] = ((uint32_t)((g >> 32) & 0x1FFFFFFull)) | (2u << 30);

        // D# group 1: data_size=4B; 1-D tensor/tile: dim0=4096, dim1=1,
        // tile_dim0=4096, tile_dim1/2 unused, dim0_stride=4096 elements.
        i32x8 g1;
        g1[0] = (int)(2u << 16);                 // workgroup_mask=0, data_size=2 (4B)
        g1[1] = (int)((uint32_t)LDIM << 16);     // tensor_dim0[15:0] @ bits 63:48
        g1[2] = (int)(1u << 16);                 // tensor_dim0 hi=0 | tensor_dim1=1
        g1[3] = (int)((uint32_t)LDIM << 16);     // tensor_dim1 hi=0 | tile_dim0=4096
        g1[4] = 0;                               // tile_dim1=0 (unused), tile_dim2=0
        g1[5] = (int)(uint32_t)LDIM;             // tensor_dim0_stride[31:0]
        g1[6] = 0;                               // stride hi | tensor_dim1_stride lo
        g1[7] = 0;

        i32x4 gz4 = {0, 0, 0, 0};                // groups 2/3 unused (<=2D tensor)
        i32x8 gz8 = {0, 0, 0, 0, 0, 0, 0, 0};    // trailing group (6-arg toolchain)

        __builtin_amdgcn_tensor_load_to_lds(g0, g1, gz4, gz4, gz8, 0);
        __builtin_amdgcn_s_wait_tensorcnt(0);    // s_wait_tensorcnt 0
    }
#else
    for (int i = threadIdx.x; i < LDIM; i += THREADS)
        lds_x[i] = src[i];
#endif
    __syncthreads();

    // ---- Stage 2: build stable sort keys: (bits(|x|) << 32) | index --------
    for (int i = threadIdx.x; i < LDIM; i += THREADS) {
        uint32_t bits = __float_as_uint(lds_x[i]) & 0x7FFFFFFFu;
        lds_key[i] = ((uint64_t)bits << 32) | (uint32_t)i;
    }
    __syncthreads();

    // ---- Stage 3: bitonic sort (ascending) of 4096 unique u64 keys ---------
    for (int k = 2; k <= LDIM; k <<= 1) {
        for (int j = k >> 1; j > 0; j >>= 1) {
            for (int i = threadIdx.x; i < LDIM; i += THREADS) {
                int ixj = i ^ j;
                if (ixj > i) {
                    uint64_t a = lds_key[i];
                    uint64_t b = lds_key[ixj];
                    bool asc = (i & k) == 0;
                    if ((a > b) == asc) {   // keys are unique (index embedded)
                        lds_key[i]   = b;
                        lds_key[ixj] = a;
                    }
                }
            }
            __syncthreads();
        }
    }

    // ---- Stage 4: scatter rank back to original index ----------------------
    for (int p = threadIdx.x; p < LDIM; p += THREADS) {
        int idx = (int)((uint32_t)lds_key[p] & 0xFFFu);
        lds_rank[idx] = (unsigned short)p;
    }
    __syncthreads();

    // ---- Stage 5: out = x + rank * sign(x) / (L-1), coalesced --------------
    float* dst = out + (size_t)row * LDIM;
    for (int i = threadIdx.x; i < LDIM; i += THREADS) {
        float v = lds_x[i];
        float s = (v > 0.0f) ? 1.0f : ((v < 0.0f) ? -1.0f : 0.0f);
        float code = ((float)lds_rank[i] * s) / (float)(LDIM - 1);
        dst[i] = v + code;
    }
}

extern "C" void kernel_launch(void* const* d_in, const int* in_sizes, int n_in,
                              void* d_out, int out_size, void* d_ws, size_t ws_size,
                              hipStream_t stream) {
    const float* x = (const float*)d_in[0];
    float* out = (float*)d_out;
    const int nrows = in_sizes[0] / LDIM;   // 8192 for the reference shapes
    rank_code_kernel<<<dim3(nrows), dim3(THREADS), 0, stream>>>(x, out, nrows);
}